// BertWithEntityStartPooling_9912784519688
// MI455X (gfx1250) — compile-verified
//
#include <hip/hip_runtime.h>
#include <float.h>

#define BATCH 64
#define SEQ   512
#define HID   1024
#define NENT  3
#define NBUF  4          // async double/quad buffering depth (4 KB per buffer)
#define NTHREADS 256     // 8 wave32s; each thread owns 4 floats of the H dim

__global__ __launch_bounds__(NTHREADS)
void entity_start_pool_kernel(const float* __restrict__ hidden,   // [B,S,H] f32
                              const int*   __restrict__ ids,      // [B,S]
                              const int*   __restrict__ amask,    // [B,S]
                              float*       __restrict__ out)      // [B,3,2H] f32
{
    __shared__ int   s_cnt[NENT];
    __shared__ int   s_list[NENT][SEQ];          // matched sequence positions
    __shared__ float stage[NBUF][NTHREADS * 4];  // 4 async-staged rows, 4 KB each

    const int b   = blockIdx.x;
    const int tid = threadIdx.x;

    if (tid < NENT) s_cnt[tid] = 0;
    __syncthreads();

    // ---- Phase 1: scan ids/mask, build per-entity position lists -----------
    for (int s = tid; s < SEQ; s += NTHREADS) {
        int id = ids[b * SEQ + s];
        int am = amask[b * SEQ + s];
        if (am != 0) {
            int e = id - 997;
            if (e >= 0 && e < NENT) {
                int slot = atomicAdd(&s_cnt[e], 1);   // ds_add_rtn_u32
                s_list[e][slot] = s;
            }
        }
    }
    __syncthreads();

    // ---- Phase 2: async-stage matched rows to LDS, max-reduce --------------
    // Thread t owns H columns [4t, 4t+3]; each lane stages & reads only its
    // own 16 bytes, so per-wave s_wait_asynccnt is the only sync needed.
    float4 acc[NENT];
    #pragma unroll
    for (int e = 0; e < NENT; ++e)
        acc[e] = make_float4(-FLT_MAX, -FLT_MAX, -FLT_MAX, -FLT_MAX);

    #pragma unroll
    for (int e = 0; e < NENT; ++e) {
        const int cnt = s_cnt[e];
        for (int base = 0; base < cnt; base += NBUF) {
            const int nb = min(NBUF, cnt - base);
            for (int j = 0; j < nb; ++j) {
                const int s = s_list[e][base + j];
                const float* src = hidden + ((size_t)b * SEQ + s) * HID + tid * 4;
                unsigned lds_addr =
                    (unsigned)(unsigned long long)(&stage[j][tid * 4]);
                unsigned long long gaddr = (unsigned long long)(size_t)src;
                // gfx1250 async tensor-path copy: 16B/lane, global -> LDS
                asm volatile("global_load_async_to_lds_b128 %0, %1, off"
                             :: "v"(lds_addr), "v"(gaddr)
                             : "memory");
            }
            asm volatile("s_wait_asynccnt 0x0" ::: "memory");
            for (int j = 0; j < nb; ++j) {
                const float4 v = *(const float4*)&stage[j][tid * 4]; // ds_load_b128
                acc[e].x = fmaxf(acc[e].x, v.x);
                acc[e].y = fmaxf(acc[e].y, v.y);
                acc[e].z = fmaxf(acc[e].z, v.z);
                acc[e].w = fmaxf(acc[e].w, v.w);
            }
        }
        if (cnt == 0) acc[e] = make_float4(0.f, 0.f, 0.f, 0.f);
    }

    // ---- Phase 3: write concatenated output (B, 3, 2H) ---------------------
    // row0 = [p0 | p1], row1 = [p0 | p2], row2 = [p1 | p2]
    float4* obase = (float4*)(out + (size_t)b * 3 * 2 * HID);
    const int q = HID / 4;             // 256 float4s per H half
    obase[0 * q + tid] = acc[0];       // row0 lo  = p0
    obase[1 * q + tid] = acc[1];       // row0 hi  = p1
    obase[2 * q + tid] = acc[0];       // row1 lo  = p0
    obase[3 * q + tid] = acc[2];       // row1 hi  = p2
    obase[4 * q + tid] = acc[1];       // row2 lo  = p1
    obase[5 * q + tid] = acc[2];       // row2 hi  = p2
}

extern "C" void kernel_launch(void* const* d_in, const int* in_sizes, int n_in,
                              void* d_out, int out_size, void* d_ws, size_t ws_size,
                              hipStream_t stream) {
    const float* hidden = (const float*)d_in[0];
    const int*   ids    = (const int*)d_in[1];
    const int*   amask  = (const int*)d_in[2];
    float*       out    = (float*)d_out;
    (void)in_sizes; (void)n_in; (void)out_size; (void)d_ws; (void)ws_size;

    entity_start_pool_kernel<<<BATCH, NTHREADS, 0, stream>>>(hidden, ids, amask, out);
}